// BatchHardTripletLoss_9079560863889
// MI455X (gfx1250) — compile-verified
//
#include <hip/hip_runtime.h>
#include <hip/hip_bf16.h>

#define NROWS 8192
#define DIMS 128
#define MARGINV 0.2f
#define EPSV 1e-6f
#define FLT_BIG 3.402823466e38f

typedef __attribute__((ext_vector_type(2))) float v2f;
typedef __attribute__((ext_vector_type(4))) float v4f;
typedef __attribute__((ext_vector_type(8))) float v8f;

#define BROWS 128   // rows per block (8 waves x 16)
#define BCOLS 64    // B-panel rows per LDS stage
#define LDST  132   // padded LDS row stride in dwords: 528B rows -> 16B aligned, conflict-free b64

// ---------------- kernel 0: per-row sum and sum-of-squares ----------------
__global__ __launch_bounds__(256) void row_stats(const float* __restrict__ x,
                                                 float* __restrict__ s,
                                                 float* __restrict__ sq) {
  const int row  = blockIdx.x * 8 + (threadIdx.x >> 5);  // one wave per row
  const int lane = threadIdx.x & 31;
  v4f v = *(const v4f*)(x + (size_t)row * DIMS + lane * 4);  // 32 lanes x 4 = 128 cols
  float sum = v[0] + v[1] + v[2] + v[3];
  float ssq = v[0]*v[0] + v[1]*v[1] + v[2]*v[2] + v[3]*v[3];
  #pragma unroll
  for (int off = 16; off > 0; off >>= 1) {
    sum += __shfl_xor(sum, off, 32);
    ssq += __shfl_xor(ssq, off, 32);
  }
  if (lane == 0) { s[row] = sum; sq[row] = ssq; }
}

// ---------- kernel 1: fused WMMA gram-tile + distance + row reductions ----------
// grid = (NROWS/BROWS, nsplit); each block reduces its column slice, writes partials.
__global__ __launch_bounds__(256) void tri_gemm(const float* __restrict__ x,
                                                const int*   __restrict__ tgt,
                                                const float* __restrict__ s,
                                                const float* __restrict__ sq,
                                                float* __restrict__ part,
                                                int nsplit) {
  __shared__ float Bp[BCOLS * LDST];
  const int tid  = threadIdx.x;
  const int wave = tid >> 5;
  const int lane = tid & 31;
  const int half = lane >> 4;   // selects K+2 for A/B, M+8 for C/D
  const int l16  = lane & 15;
  const int rowBase = blockIdx.x * BROWS + wave * 16;
  const int split = blockIdx.y;
  const int colsPerSplit = NROWS / nsplit;
  const int jLo = split * colsPerSplit;
  const int jHi = jLo + colsPerSplit;

  // A tile (16 rows x 128 K) resident in VGPRs, ISA 16x4xf32 A-layout per k-step.
  v2f a[32];
  {
    const float* arow = x + (size_t)(rowBase + l16) * DIMS + 2 * half;
    #pragma unroll
    for (int ks = 0; ks < 32; ++ks) a[ks] = *(const v2f*)(arow + 4 * ks);
  }
  // Row-side scalars for this lane's 8 M slots (C/D layout: slot v -> M = v + 8*half).
  float sr[8], sqr[8]; int tr[8];
  #pragma unroll
  for (int v = 0; v < 8; ++v) {
    const int m = rowBase + v + 8 * half;
    sr[v] = s[m]; sqr[v] = sq[m]; tr[v] = tgt[m];
  }
  float hp[8], mn[8], ms[8], ma[8];  // hardest-pos, min-neg, min-same, max-all
  #pragma unroll
  for (int v = 0; v < 8; ++v) { hp[v] = 0.f; mn[v] = FLT_BIG; ms[v] = FLT_BIG; ma[v] = 0.f; }

  for (int j0 = jLo; j0 < jHi; j0 += BCOLS) {
    __syncthreads();  // previous panel fully consumed
    // Cooperative B-panel stage: each wave streams full 512B rows (coalesced b128).
    #pragma unroll
    for (int rr = 0; rr < BCOLS / 8; ++rr) {
      const int r = rr * 8 + wave;
      v4f val = *(const v4f*)(x + (size_t)(j0 + r) * DIMS + lane * 4);
      *(v4f*)(&Bp[r * LDST + lane * 4]) = val;
    }
    if (j0 + BCOLS < jHi)  // pull next panel toward L2/L0 (global_prefetch_b8)
      __builtin_prefetch(x + (size_t)(j0 + BCOLS + (tid >> 2)) * DIMS + (size_t)(tid & 3) * 32, 0, 0);
    __syncthreads();

    #pragma unroll
    for (int jj = 0; jj < BCOLS / 16; ++jj) {
      v8f c = {0.f, 0.f, 0.f, 0.f, 0.f, 0.f, 0.f, 0.f};
      const float* brow = &Bp[(jj * 16 + l16) * LDST + 2 * half];
      #pragma unroll
      for (int ks = 0; ks < 32; ++ks) {
        v2f b = *(const v2f*)(brow + 4 * ks);  // ds_load_b64, conflict-free
        c = __builtin_amdgcn_wmma_f32_16x16x4_f32(false, a[ks], false, b,
                                                  (short)0, c, false, false);
      }
      // Epilogue: tile of dot -> distance -> running row reductions (C/D layout).
      const int col = j0 + jj * 16 + l16;
      const float sc = s[col], sqc = sq[col];
      const int tc = tgt[col];
      #pragma unroll
      for (int v = 0; v < 8; ++v) {
        const int row = rowBase + v + 8 * half;
        float d2 = sqc + sqr[v] - 2.f * c[v]
                 + 2.f * EPSV * (sc - sr[v]) + (float)DIMS * EPSV * EPSV;
        float d = sqrtf(fmaxf(d2, 0.f));
        const bool same = (tc == tr[v]);
        ma[v] = fmaxf(ma[v], d);
        hp[v] = fmaxf(hp[v], (same && (col != row)) ? d : 0.f);
        mn[v] = fminf(mn[v], same ? FLT_BIG : d);
        ms[v] = fminf(ms[v], same ? d : FLT_BIG);
      }
    }
  }

  // Finish reductions across the 16 lanes sharing each M (xor stays within half).
  #pragma unroll
  for (int v = 0; v < 8; ++v) {
    #pragma unroll
    for (int off = 1; off < 16; off <<= 1) {
      ma[v] = fmaxf(ma[v], __shfl_xor(ma[v], off, 32));
      hp[v] = fmaxf(hp[v], __shfl_xor(hp[v], off, 32));
      mn[v] = fminf(mn[v], __shfl_xor(mn[v], off, 32));
      ms[v] = fminf(ms[v], __shfl_xor(ms[v], off, 32));
    }
  }
  if (l16 == 0) {  // lane 0 -> rows M=0..7, lane 16 -> rows M=8..15
    const size_t q = (size_t)NROWS * nsplit;
    #pragma unroll
    for (int v = 0; v < 8; ++v) {
      const int row = rowBase + v + 8 * half;
      float* p = part + (size_t)row * nsplit + split;
      p[0 * q] = hp[v]; p[1 * q] = mn[v]; p[2 * q] = ms[v]; p[3 * q] = ma[v];
    }
  }
}

// ---------------- kernel 2: merge column-split partials -> per-row loss ----------------
__global__ __launch_bounds__(256) void tri_combine(const float* __restrict__ part,
                                                   float* __restrict__ loss,
                                                   int nsplit) {
  const int row = blockIdx.x * 256 + threadIdx.x;
  const size_t q = (size_t)NROWS * nsplit;
  float hp = 0.f, mn = FLT_BIG, ms = FLT_BIG, ma = 0.f;
  for (int sp = 0; sp < nsplit; ++sp) {
    const size_t i = (size_t)row * nsplit + sp;
    hp = fmaxf(hp, part[0 * q + i]);
    mn = fminf(mn, part[1 * q + i]);
    ms = fminf(ms, part[2 * q + i]);
    ma = fmaxf(ma, part[3 * q + i]);
  }
  // hardest_neg = min( min_neg , max_all + min_same )  (reference's push-up trick)
  const float hn = fminf(mn, ma + ms);
  loss[row] = fmaxf(hp - hn + MARGINV, 0.f);
}

// ---------------- kernel 3: deterministic mean ----------------
__global__ __launch_bounds__(256) void reduce_mean(const float* __restrict__ loss,
                                                   float* __restrict__ out) {
  __shared__ float sm[256];
  float acc = 0.f;
  for (int i = threadIdx.x; i < NROWS; i += 256) acc += loss[i];
  sm[threadIdx.x] = acc;
  __syncthreads();
  for (int s2 = 128; s2 > 0; s2 >>= 1) {
    if ((int)threadIdx.x < s2) sm[threadIdx.x] += sm[threadIdx.x + s2];
    __syncthreads();
  }
  if (threadIdx.x == 0) out[0] = sm[0] / (float)NROWS;
}

extern "C" void kernel_launch(void* const* d_in, const int* in_sizes, int n_in,
                              void* d_out, int out_size, void* d_ws, size_t ws_size,
                              hipStream_t stream) {
  (void)in_sizes; (void)n_in; (void)out_size;
  const float* x   = (const float*)d_in[0];
  const int*   tgt = (const int*)d_in[1];
  float* ws   = (float*)d_ws;
  float* s    = ws;
  float* sq   = ws + NROWS;
  float* loss = ws + 2 * NROWS;
  float* part = ws + 3 * NROWS;

  int nsplit = 8;  // 512 blocks to cover the WGP array; fall back if ws is tight
  size_t need = ((size_t)3 * NROWS + (size_t)4 * NROWS * nsplit) * sizeof(float);
  if (ws_size < need) nsplit = 1;

  row_stats<<<NROWS / 8, 256, 0, stream>>>(x, s, sq);
  dim3 grid(NROWS / BROWS, nsplit);
  tri_gemm<<<grid, 256, 0, stream>>>(x, tgt, s, sq, part, nsplit);
  tri_combine<<<NROWS / 256, 256, 0, stream>>>(part, loss, nsplit);
  reduce_mean<<<1, 256, 0, stream>>>(loss, (float*)d_out);
}